// HeteroGATConv_7086696038634
// MI455X (gfx1250) — compile-verified
//
#include <hip/hip_runtime.h>
#include <hip/hip_bf16.h>

typedef __attribute__((ext_vector_type(2))) float v2f;
typedef __attribute__((ext_vector_type(8))) float v8f;

#define IN_C   128
#define HC     128   // heads*out_c = 2*64
#define OUT_C  64
#define NEG_SLOPE 0.2f

#define ROWS_PER_BLK 32
#define XS_STRIDE 130   // pad 128 -> 130 floats: breaks 512B bank stride, keeps 8B align

// ---------------- helpers ----------------

__device__ __forceinline__ unsigned encodeOrdered(float f) {
    unsigned u = __float_as_uint(f);
    return (u & 0x80000000u) ? ~u : (u | 0x80000000u);
}
__device__ __forceinline__ float decodeOrdered(unsigned u) {
    unsigned b = (u & 0x80000000u) ? (u ^ 0x80000000u) : ~u;
    return __uint_as_float(b);
}
__device__ __forceinline__ float lrelu(float x) {
    return x > 0.0f ? x : x * NEG_SLOPE;
}

// ---------------- init kernels ----------------

__global__ void fill_f32(float* __restrict__ p, float v, long n) {
    long i = (long)blockIdx.x * blockDim.x + threadIdx.x;
    if (i < n) p[i] = v;
}
__global__ void fill_u32(unsigned* __restrict__ p, unsigned v, long n) {
    long i = (long)blockIdx.x * blockDim.x + threadIdx.x;
    if (i < n) p[i] = v;
}

// ---------------- projection: Y[N,128] = X[N,128] @ W[128,128] + b ----------------
// Block = 256 threads = 8 waves. Block computes a 32x128 output tile.
// X tile (32 rows, clamped) is staged in LDS once with coalesced b128 loads;
// each wave owns 16 output columns and two 16-row accumulators, chaining
// 32x V_WMMA_F32_16X16X4_F32 (K=128). B pair comes straight from global
// (W is 64KB, L2-resident) and is reused by both row-tiles.
__global__ __launch_bounds__(256) void gemm_bias_wmma(
    const float* __restrict__ X, const float* __restrict__ W,
    const float* __restrict__ bias, float* __restrict__ Y, int N)
{
    __shared__ float Xs[ROWS_PER_BLK * XS_STRIDE];   // 16640 B

    const int tid  = threadIdx.x;
    const int lane = tid & 31;
    const int wave = tid >> 5;                       // column tile 0..7
    const int row0 = blockIdx.x * ROWS_PER_BLK;

    // ---- stage X rows into LDS (row index clamped; stores are guarded later) ----
    for (int idx = tid; idx < ROWS_PER_BLK * (IN_C / 4); idx += 256) {
        int r  = idx >> 5;                // 0..31
        int c4 = idx & 31;                // float4 index within row
        int row = row0 + r;
        if (row >= N) row = N - 1;
        float4 x = ((const float4*)(X + (size_t)row * IN_C))[c4];
        float* p = &Xs[r * XS_STRIDE + c4 * 4];
        p[0] = x.x; p[1] = x.y; p[2] = x.z; p[3] = x.w;
    }
    __syncthreads();

    const int m    = lane & 15;
    const int half = lane >> 4;                      // 0 / 1
    const int kb   = half * 2;
    const int col  = wave * 16 + m;

    v8f acc0 = {};
    v8f acc1 = {};
#pragma unroll
    for (int k0 = 0; k0 < IN_C; k0 += 4) {
        v2f b;
        b.x = W[(size_t)(k0 + kb)     * HC + col];
        b.y = W[(size_t)(k0 + kb + 1) * HC + col];
        v2f a0 = *(const v2f*)&Xs[m        * XS_STRIDE + k0 + kb];
        v2f a1 = *(const v2f*)&Xs[(m + 16) * XS_STRIDE + k0 + kb];
        acc0 = __builtin_amdgcn_wmma_f32_16x16x4_f32(false, a0, false, b,
                                                     (short)0, acc0, false, false);
        acc1 = __builtin_amdgcn_wmma_f32_16x16x4_f32(false, a1, false, b,
                                                     (short)0, acc1, false, false);
    }

    const float bcol = bias[col];
#pragma unroll
    for (int r = 0; r < 8; ++r) {
        int orow0 = row0 + r + half * 8;
        if (orow0 < N) Y[(size_t)orow0 * HC + col] = acc0[r] + bcol;
        int orow1 = row0 + 16 + r + half * 8;
        if (orow1 < N) Y[(size_t)orow1 * HC + col] = acc1[r] + bcol;
    }
}

// ---------------- pass B: edge logits + segment max ----------------
// One wave per edge. Lane l covers packed channels [4l,4l+3] of the 128-dim
// (head = l>>4, in-head channel = (l&15)*4). Half-wave xor-reduce -> per-head logit.
__global__ __launch_bounds__(256) void edge_logits_max(
    const float* __restrict__ xl, const float* __restrict__ xr,
    const int* __restrict__ src, const int* __restrict__ dst,
    const float* __restrict__ att,        // [2,64]
    float* __restrict__ logits,           // [E,2]
    unsigned* __restrict__ mmax,          // [Ndst,2] ordered-encoded
    int E)
{
    int e = blockIdx.x * 8 + (threadIdx.x >> 5);
    if (e >= E) return;
    const int lane = threadIdx.x & 31;
    const int s = src[e], d = dst[e];

    const float4 xj = ((const float4*)(xl + (size_t)s * HC))[lane];
    const float4 xi = ((const float4*)(xr + (size_t)d * HC))[lane];
    const float4 at = ((const float4*)att)[lane];

    float v = lrelu(xi.x + xj.x) * at.x
            + lrelu(xi.y + xj.y) * at.y
            + lrelu(xi.z + xj.z) * at.z
            + lrelu(xi.w + xj.w) * at.w;

    // reduce within each 16-lane half (one head each)
    v += __shfl_xor(v, 8);
    v += __shfl_xor(v, 4);
    v += __shfl_xor(v, 2);
    v += __shfl_xor(v, 1);

    if ((lane & 15) == 0) {
        int h = lane >> 4;
        logits[(size_t)e * 2 + h] = v;
        atomicMax(&mmax[(size_t)d * 2 + h], encodeOrdered(v));
    }
}

// ---------------- pass C: a = exp(logit - m); denom += a ----------------
__global__ void edge_exp_sum(
    const int* __restrict__ dst,
    float* __restrict__ logits,           // in: logits, out: a
    const unsigned* __restrict__ mmax,
    float* __restrict__ denom,            // [Ndst,2]
    int E)
{
    int i = blockIdx.x * blockDim.x + threadIdx.x;   // over E*2
    if (i >= E * 2) return;
    int e = i >> 1, h = i & 1;
    int d = dst[e];
    float m = decodeOrdered(mmax[(size_t)d * 2 + h]);
    float a = __expf(logits[i] - m);
    logits[i] = a;
    atomicAdd(&denom[(size_t)d * 2 + h], a);
}

// ---------------- pass D: out[d] += 0.5 * alpha * xj (head-mean folded) ----------------
__global__ __launch_bounds__(256) void edge_scatter(
    const float* __restrict__ xl,
    const int* __restrict__ src, const int* __restrict__ dst,
    const float* __restrict__ a,          // [E,2]
    const float* __restrict__ denom,      // [Ndst,2]
    float* __restrict__ out,              // [Ndst,64]
    int E)
{
    int e = blockIdx.x * 8 + (threadIdx.x >> 5);
    if (e >= E) return;
    const int lane = threadIdx.x & 31;
    const int s = src[e], d = dst[e];
    const int h = lane >> 4;

    float alpha = a[(size_t)e * 2 + h] / (denom[(size_t)d * 2 + h] + 1e-16f);
    float w = 0.5f * alpha;

    const float4 xj = ((const float4*)(xl + (size_t)s * HC))[lane];
    float* o = out + (size_t)d * OUT_C + (lane & 15) * 4;
    atomicAdd(o + 0, w * xj.x);
    atomicAdd(o + 1, w * xj.y);
    atomicAdd(o + 2, w * xj.z);
    atomicAdd(o + 3, w * xj.w);
}

// ---------------- pass E: bias add ----------------
__global__ void bias_add(float* __restrict__ out, const float* __restrict__ bias, int N) {
    long i = (long)blockIdx.x * blockDim.x + threadIdx.x;
    if (i < (long)N * OUT_C) out[i] += bias[i & (OUT_C - 1)];
}

// ---------------- launch ----------------

extern "C" void kernel_launch(void* const* d_in, const int* in_sizes, int n_in,
                              void* d_out, int out_size, void* d_ws, size_t ws_size,
                              hipStream_t stream) {
    const float* x_user  = (const float*)d_in[0];
    const float* x_item  = (const float*)d_in[1];
    const int*   src_u2i = (const int*)d_in[2];
    const int*   dst_u2i = (const int*)d_in[3];
    const int*   src_i2u = (const int*)d_in[4];
    const int*   dst_i2u = (const int*)d_in[5];
    const float* Wl_u2i  = (const float*)d_in[6];
    const float* bl_u2i  = (const float*)d_in[7];
    const float* Wr_u2i  = (const float*)d_in[8];
    const float* br_u2i  = (const float*)d_in[9];
    const float* att_u2i = (const float*)d_in[10];
    const float* bias_u2i= (const float*)d_in[11];
    const float* Wl_i2u  = (const float*)d_in[12];
    const float* bl_i2u  = (const float*)d_in[13];
    const float* Wr_i2u  = (const float*)d_in[14];
    const float* br_i2u  = (const float*)d_in[15];
    const float* att_i2u = (const float*)d_in[16];
    const float* bias_i2u= (const float*)d_in[17];

    const int NU = in_sizes[0] / IN_C;
    const int NI = in_sizes[1] / IN_C;
    const int E  = in_sizes[2];

    // workspace layout (floats)
    float* ws = (float*)d_ws;
    size_t off = 0;
    float* xl_u2i = ws + off; off += (size_t)NU * HC;   // user src proj
    float* xr_u2i = ws + off; off += (size_t)NI * HC;   // item dst proj
    float* xl_i2u = ws + off; off += (size_t)NI * HC;   // item src proj
    float* xr_i2u = ws + off; off += (size_t)NU * HC;   // user dst proj
    float* lg_u2i = ws + off; off += (size_t)E * 2;     // logits -> a
    float* lg_i2u = ws + off; off += (size_t)E * 2;
    unsigned* m_u2i = (unsigned*)(ws + off); off += (size_t)NI * 2;
    unsigned* m_i2u = (unsigned*)(ws + off); off += (size_t)NU * 2;
    float* dn_u2i = ws + off; off += (size_t)NI * 2;
    float* dn_i2u = ws + off; off += (size_t)NU * 2;

    float* out_user = (float*)d_out;                    // [NU,64]
    float* out_item = (float*)d_out + (size_t)NU * OUT_C;

    // init accumulators / maxima / output
    {
        long n = (long)NI * 2;
        fill_u32<<<(n + 255) / 256, 256, 0, stream>>>(m_u2i, 0u, n);
        fill_f32<<<(n + 255) / 256, 256, 0, stream>>>(dn_u2i, 0.0f, n);
        n = (long)NU * 2;
        fill_u32<<<(n + 255) / 256, 256, 0, stream>>>(m_i2u, 0u, n);
        fill_f32<<<(n + 255) / 256, 256, 0, stream>>>(dn_i2u, 0.0f, n);
        n = (long)(NU + NI) * OUT_C;
        fill_f32<<<(n + 255) / 256, 256, 0, stream>>>((float*)d_out, 0.0f, n);
    }

    // projections (f32 WMMA)
    const int gu = (NU + ROWS_PER_BLK - 1) / ROWS_PER_BLK;
    const int gi = (NI + ROWS_PER_BLK - 1) / ROWS_PER_BLK;
    gemm_bias_wmma<<<gu, 256, 0, stream>>>(x_user, Wl_u2i, bl_u2i, xl_u2i, NU);
    gemm_bias_wmma<<<gi, 256, 0, stream>>>(x_item, Wr_u2i, br_u2i, xr_u2i, NI);
    gemm_bias_wmma<<<gi, 256, 0, stream>>>(x_item, Wl_i2u, bl_i2u, xl_i2u, NI);
    gemm_bias_wmma<<<gu, 256, 0, stream>>>(x_user, Wr_i2u, br_i2u, xr_i2u, NU);

    const int egrid = (E + 7) / 8;

    // direction u2i (dst = item)
    edge_logits_max<<<egrid, 256, 0, stream>>>(xl_u2i, xr_u2i, src_u2i, dst_u2i,
                                               att_u2i, lg_u2i, m_u2i, E);
    edge_exp_sum<<<(E * 2 + 255) / 256, 256, 0, stream>>>(dst_u2i, lg_u2i, m_u2i, dn_u2i, E);
    edge_scatter<<<egrid, 256, 0, stream>>>(xl_u2i, src_u2i, dst_u2i, lg_u2i, dn_u2i, out_item, E);
    bias_add<<<((long)NI * OUT_C + 255) / 256, 256, 0, stream>>>(out_item, bias_u2i, NI);

    // direction i2u (dst = user)
    edge_logits_max<<<egrid, 256, 0, stream>>>(xl_i2u, xr_i2u, src_i2u, dst_i2u,
                                               att_i2u, lg_i2u, m_i2u, E);
    edge_exp_sum<<<(E * 2 + 255) / 256, 256, 0, stream>>>(dst_i2u, lg_i2u, m_i2u, dn_i2u, E);
    edge_scatter<<<egrid, 256, 0, stream>>>(xl_i2u, src_i2u, dst_i2u, lg_i2u, dn_i2u, out_user, E);
    bias_add<<<((long)NU * OUT_C + 255) / 256, 256, 0, stream>>>(out_user, bias_i2u, NU);

    (void)n_in; (void)out_size; (void)ws_size;
}